// nvilt2_2869038154224
// MI455X (gfx1250) — compile-verified
//
#include <hip/hip_runtime.h>
#include <math.h>

// ---------------------------------------------------------------------------
// Inverse-litho pipeline for gfx1250 (MI455X), compile-only target.
//
// Key algebra (see analysis): both FFTs are partial DFTs over a 35x35
// frequency window; x_out_max = sigmoid(50*(1.02^2*aerial_kf - th));
// dose 0.98 and sqrt(scale_k)/D^2 are folded into the per-k spectral matrix.
// Dominant GEMM (img = B * Wx^T, M=N=620, K=35->36, complex, 192 passes)
// runs on V_WMMA_F32_16X16X4_F32.
//
// Workspace requirement: ~41.4 MB (asserted by layout below).
// ---------------------------------------------------------------------------

#define DN    620
#define KDN   35
#define KCNT  24
#define NB    4
#define OFFI  293            // absolute freq start for the inverse transform
#define FWDF  603            // (OFF + D/2) % D : forward-window freq base
#define NTOT  (NB*DN*DN)     // elements per output tensor (1,537,600)
#define KPAD  36             // contraction dim padded to multiple of 4
#define YPAD  624            // 39*16 rows for 16x16 tiling
#define PLANE (YPAD*KPAD)    // one re or im plane of B per (n,set,k)

// workspace layout (float offsets)
#define WS_TAB  0                              // 620 cos + 620 sin
#define WS_MASK (WS_TAB + 2*DN)                // N*D*D
#define WS_T1   (WS_MASK + NB*DN*DN)           // N*35*620 complex (interleaved)
#define WS_WIN  (WS_T1 + NB*KDN*DN*2)          // N*35*35 complex (interleaved)
#define WS_B    (WS_WIN + NB*KDN*KDN*2)        // 192 * 2 planes of 624*36

typedef __attribute__((ext_vector_type(2))) float v2f;
typedef __attribute__((ext_vector_type(8))) float v8f;

static __device__ __forceinline__ v8f wmma4(v2f a, v2f b, v8f c) {
  // D = A(16x4, f32) * B(4x16, f32) + C(16x16, f32)
  return __builtin_amdgcn_wmma_f32_16x16x4_f32(false, a, false, b,
                                               (short)0, c, false, false);
}

static __device__ __forceinline__ float sigf(float z) {
  return 1.0f / (1.0f + __expf(-z));
}

// --- K0: sincos table: ctab[i]=cos(2*pi*i/620), stab[i]=sin(...) ------------
__global__ void nvilt_k0_tab(float* __restrict__ ws) {
  int i = blockIdx.x * blockDim.x + threadIdx.x;
  if (i >= DN) return;
  float th = 6.28318530717958647692f * (float)i / (float)DN;
  ws[WS_TAB + i]      = cosf(th);
  ws[WS_TAB + DN + i] = sinf(th);
}

// --- K1: 3x3 zero-padded avg pool + mask sigmoid ----------------------------
__global__ void nvilt_k1_mask(const float* __restrict__ x, float* __restrict__ ws) {
  int t = blockIdx.x * blockDim.x + threadIdx.x;
  if (t >= NTOT) return;
  int xx = t % DN;
  int yy = (t / DN) % DN;
  int n  = t / (DN * DN);
  const float* p = x + (size_t)n * DN * DN;
  float s = 0.0f;
#pragma unroll
  for (int dy = -1; dy <= 1; ++dy)
#pragma unroll
    for (int dx = -1; dx <= 1; ++dx) {
      int Y = yy + dy, X = xx + dx;
      if (Y >= 0 && Y < DN && X >= 0 && X < DN) s += p[(size_t)Y * DN + X];
    }
  float m = s * (1.0f / 9.0f);
  ws[WS_MASK + t] = sigf(4.0f * (m - 0.5f));
}

// --- K2a: partial column DFT: T1[n,r,x] = sum_y mask[n,y,x] e^{-2pi i f y/D}
__global__ void nvilt_k2a(float* __restrict__ ws) {
  int t = blockIdx.x * blockDim.x + threadIdx.x;
  if (t >= NB * KDN * DN) return;
  int xx = t % DN;
  int r  = (t / DN) % KDN;
  int n  = t / (DN * KDN);
  const float* ct = ws + WS_TAB;
  const float* st = ct + DN;
  const float* m  = ws + WS_MASK + (size_t)n * DN * DN + xx;
  int f = (FWDF + r) % DN;
  float re = 0.0f, im = 0.0f;
  int idx = 0;
  for (int y = 0; y < DN; ++y) {
    float v = m[(size_t)y * DN];
    re += v * ct[idx];
    im -= v * st[idx];
    idx += f; if (idx >= DN) idx -= DN;
  }
  float* o = ws + WS_T1 + (size_t)((n * KDN + r) * DN + xx) * 2;
  o[0] = re; o[1] = im;
}

// --- K2b: partial row DFT: win[n,r,c] = sum_x T1[n,r,x] e^{-2pi i f x/D} ----
__global__ void nvilt_k2b(float* __restrict__ ws) {
  int t = blockIdx.x * blockDim.x + threadIdx.x;
  if (t >= NB * KDN * KDN) return;
  int c = t % KDN;
  // decode matches storage: t == (n*KDN + r)*KDN + c
  const float* ct = ws + WS_TAB;
  const float* st = ct + DN;
  int r = (t / KDN) % KDN;
  int n = t / (KDN * KDN);
  int f = (FWDF + c) % DN;
  const float* T = ws + WS_T1 + (size_t)(n * KDN + r) * DN * 2;
  float re = 0.0f, im = 0.0f;
  int idx = 0;
  for (int xx = 0; xx < DN; ++xx) {
    float tr = T[xx * 2], ti = T[xx * 2 + 1];
    float cc = ct[idx], ss = st[idx];
    re += tr * cc + ti * ss;
    im += ti * cc - tr * ss;
    idx += f; if (idx >= DN) idx -= DN;
  }
  float* o = ws + WS_WIN + (size_t)t * 2;
  o[0] = re; o[1] = im;
}

// --- K3: B[n,set,k][y,c] = sum_r e^{+2pi i (293+r)y/D} * g * (win[n] o ker)[r,c]
// g folds dose (set1: 0.98), sqrt(scale_k) and 1/D^2.
__global__ void nvilt_k3_buildB(float* __restrict__ ws,
    const float* __restrict__ kf_re,  const float* __restrict__ kf_im,
    const float* __restrict__ kdf_re, const float* __restrict__ kdf_im,
    const float* __restrict__ kf_sc,  const float* __restrict__ kdf_sc) {
  int t = blockIdx.x * blockDim.x + threadIdx.x;
  int c = t % KPAD;  t /= KPAD;
  int y = t % YPAD;  t /= YPAD;
  int k = t % KCNT;  t /= KCNT;
  int set = t & 1;
  int n   = t >> 1;

  float* Bre = ws + WS_B + (size_t)((n * 2 + set) * KCNT + k) * (2 * PLANE) + (size_t)y * KPAD;
  float* Bim = Bre + PLANE;
  if (c >= KDN) { Bre[c] = 0.0f; Bim[c] = 0.0f; return; }  // K-dim zero pad

  const float* ct  = ws + WS_TAB;
  const float* st  = ct + DN;
  const float* kre = (set ? kdf_re : kf_re) + (size_t)k * KDN * KDN + c;
  const float* kim = (set ? kdf_im : kf_im) + (size_t)k * KDN * KDN + c;
  const float* w   = ws + WS_WIN + (size_t)(n * KDN * KDN + c) * 2;
  float sc = (set ? kdf_sc : kf_sc)[k];
  float g  = (set ? 0.98f : 1.0f) * sqrtf(sc) * (1.0f / ((float)DN * (float)DN));

  int idx  = (OFFI * y) % DN;   // phase (293+r)*y, r=0
  int step = y % DN;
  float bre = 0.0f, bim = 0.0f;
  for (int r = 0; r < KDN; ++r) {
    float wrr = w[(size_t)r * KDN * 2];
    float wii = w[(size_t)r * KDN * 2 + 1];
    float krr = kre[(size_t)r * KDN];
    float kii = kim[(size_t)r * KDN];
    float are = g * (wrr * krr - wii * kii);
    float aim = g * (wrr * kii + wii * krr);
    float cc = ct[idx], ss = st[idx];
    bre += cc * are - ss * aim;
    bim += cc * aim + ss * are;
    idx += step; if (idx >= DN) idx -= DN;
  }
  Bre[c] = bre; Bim[c] = bim;
}

// --- K4: WMMA GEMM img = B * Wx^T per k, accumulate aerial = sum_k |img|^2 --
// One wave per 16x16 output tile. Complex GEMM via 4 f32 accumulator chains.
__global__ __launch_bounds__(32) void nvilt_k4_aerial(
    const float* __restrict__ ws, float* __restrict__ out) {
  const float* ct = ws + WS_TAB;
  const float* st = ct + DN;
  const float* Bbase = ws + WS_B;

  int tile = blockIdx.x;
  int tx = tile % 39; tile /= 39;
  int ty = tile % 39; tile /= 39;
  int set = tile & 1;
  int n   = tile >> 1;

  int lane   = threadIdx.x;
  int laneM  = lane & 15;
  int laneHi = lane >> 4;
  int x0 = tx * 16, y0 = ty * 16;
  int xcol = x0 + laneM;          // N column for B-operand / C tile
  int yrow = y0 + laneM;          // M row for A-operand
  int aoff = yrow * KPAD;

  // Precompute Wx^T operand values (B-operand layout: lanes<16 hold K-pair
  // {4kk, 4kk+1}, lanes>=16 hold {4kk+2, 4kk+3}, N = lane%16).
  v2f wr[9], wi[9];
#pragma unroll
  for (int kk = 0; kk < 9; ++kk) {
    int c0 = 4 * kk + 2 * laneHi;
    int c1 = c0 + 1;
    int i0 = ((OFFI + c0) * xcol) % DN;
    int i1 = ((OFFI + c1) * xcol) % DN;
    float cr0 = (c0 < KDN) ? ct[i0] : 0.0f;
    float ci0 = (c0 < KDN) ? st[i0] : 0.0f;
    float cr1 = (c1 < KDN) ? ct[i1] : 0.0f;
    float ci1 = (c1 < KDN) ? st[i1] : 0.0f;
    v2f a; a.x = cr0; a.y = cr1; wr[kk] = a;
    v2f b; b.x = ci0; b.y = ci1; wi[kk] = b;
  }

  v8f acc = {};
  const float* nb = Bbase + (size_t)((n * 2 + set) * KCNT) * (2 * PLANE);
  for (int k = 0; k < KCNT; ++k) {
    const float* Bre = nb + (size_t)k * (2 * PLANE);
    const float* Bim = Bre + PLANE;
    v8f P = {}, Q = {}, R = {}, S = {};
#pragma unroll
    for (int kk = 0; kk < 9; ++kk) {
      int c0 = 4 * kk + 2 * laneHi;   // A-operand K-pair (same split as B)
      v2f ar = *(const v2f*)(Bre + aoff + c0);
      v2f ai = *(const v2f*)(Bim + aoff + c0);
      P = wmma4(ar, wr[kk], P);       // Br*Wr
      Q = wmma4(ai, wi[kk], Q);       // Bi*Wi
      R = wmma4(ar, wi[kk], R);       // Br*Wi
      S = wmma4(ai, wr[kk], S);       // Bi*Wr
    }
#pragma unroll
    for (int j = 0; j < 8; ++j) {
      float re = P[j] - Q[j];
      float im = R[j] + S[j];
      acc[j] += re * re + im * im;
    }
  }

  // C/D layout: lanes<16: VGPR j -> M=j; lanes>=16: M=8+j; N = lane%16.
  if (xcol < DN) {
    float* o = out + (set ? (size_t)2 * NTOT : (size_t)0) + (size_t)n * DN * DN;
#pragma unroll
    for (int j = 0; j < 8; ++j) {
      int y = y0 + 8 * laneHi + j;
      if (y < DN) o[(size_t)y * DN + xcol] = acc[j];
    }
  }
}

// --- K5: resist sigmoids; x_out_max from 1.02^2 * aerial_kf (dose identity) -
__global__ void nvilt_k5_sig(float* __restrict__ out) {
  int t = blockIdx.x * blockDim.x + threadIdx.x;
  if (t >= NTOT) return;
  float akf  = out[t];                       // aerial (kf, dose 1.0)
  float akdf = out[t + (size_t)2 * NTOT];    // aerial (kdf, 0.98 folded in)
  out[t]                      = sigf(50.0f * (akf - 0.225f));
  out[t + (size_t)NTOT]       = sigf(50.0f * (akf * 1.0404f - 0.225f));
  out[t + (size_t)2 * NTOT]   = sigf(50.0f * (akdf - 0.225f));
}

extern "C" void kernel_launch(void* const* d_in, const int* in_sizes, int n_in,
                              void* d_out, int out_size, void* d_ws, size_t ws_size,
                              hipStream_t stream) {
  (void)in_sizes; (void)n_in; (void)out_size; (void)ws_size;
  const float* x       = (const float*)d_in[0];
  const float* kf_re   = (const float*)d_in[1];
  const float* kf_im   = (const float*)d_in[2];
  const float* kdf_re  = (const float*)d_in[3];
  const float* kdf_im  = (const float*)d_in[4];
  const float* kf_sc   = (const float*)d_in[5];
  const float* kdf_sc  = (const float*)d_in[6];
  float* out = (float*)d_out;
  float* ws  = (float*)d_ws;   // needs >= (WS_B + 192*2*PLANE)*4 ~ 41.4 MB

  nvilt_k0_tab <<<(DN + 255) / 256, 256, 0, stream>>>(ws);
  nvilt_k1_mask<<<(NTOT + 255) / 256, 256, 0, stream>>>(x, ws);
  nvilt_k2a    <<<(NB * KDN * DN + 255) / 256, 256, 0, stream>>>(ws);
  nvilt_k2b    <<<(NB * KDN * KDN + 255) / 256, 256, 0, stream>>>(ws);
  nvilt_k3_buildB<<<(NB * 2 * KCNT * YPAD * KPAD) / 256, 256, 0, stream>>>(
      ws, kf_re, kf_im, kdf_re, kdf_im, kf_sc, kdf_sc);
  nvilt_k4_aerial<<<NB * 2 * 39 * 39, 32, 0, stream>>>(ws, out);
  nvilt_k5_sig <<<(NTOT + 255) / 256, 256, 0, stream>>>(out);
}